// GAT_65730179498119
// MI455X (gfx1250) — compile-verified
//
#include <hip/hip_runtime.h>
#include <cmath>

typedef __attribute__((ext_vector_type(2))) float v2f;
typedef __attribute__((ext_vector_type(8))) float v8f;

#define NEG_SLOPE 0.2f

// ---- monotonic float<->uint encoding so atomicMax(u32) == float max ----
__device__ __forceinline__ unsigned fenc(float f) {
  unsigned u = __float_as_uint(f);
  return (u & 0x80000000u) ? ~u : (u | 0x80000000u);
}
__device__ __forceinline__ float fdec(unsigned u) {
  return (u & 0x80000000u) ? __uint_as_float(u & 0x7FFFFFFFu)
                           : __uint_as_float(~u);
}

__device__ __forceinline__ float lrelu(float v) {
  return v > 0.0f ? v : NEG_SLOPE * v;
}

// =====================================================================
// Layer-1 GEMM: h1[N,64] = x[N,128] @ W1[128,64] via V_WMMA_F32_16X16X4_F32
// 4 waves / block; wave w owns column tile [16w,16w+16) (== head w).
// A frag: m = lane&15, k = vgpr + 2*(lane>>4).  B frag: n = lane&15, same k.
// C frag: col = lane&15, row = vgpr + 8*(lane>>4).
// =====================================================================
__global__ void gemm1_wmma(const float* __restrict__ x,
                           const float* __restrict__ W1,
                           float* __restrict__ h1) {
  const int wave = threadIdx.x >> 5;          // 0..3
  const int lane = threadIdx.x & 31;
  const int m    = lane & 15;
  const int kh   = lane >> 4;                 // 0 or 1
  const int row0 = blockIdx.x * 16;
  const int col0 = wave * 16;

  v8f acc = {};
  const float* xrow = x + (size_t)(row0 + m) * 128;
  for (int k0 = 0; k0 < 128; k0 += 4) {
    const int ka = k0 + 2 * kh;
    v2f a;
    a.x = xrow[ka];
    a.y = xrow[ka + 1];
    v2f b;
    b.x = W1[(size_t)ka       * 64 + col0 + m];
    b.y = W1[(size_t)(ka + 1) * 64 + col0 + m];
    acc = __builtin_amdgcn_wmma_f32_16x16x4_f32(false, a, false, b,
                                                (short)0, acc, false, false);
  }
#pragma unroll
  for (int r = 0; r < 8; ++r) {
    const int row = row0 + r + 8 * kh;
    h1[(size_t)row * 64 + col0 + m] = acc[r];
  }
}

// =====================================================================
// Layer-2 GEMM: h2[N,16] = emb[N,64] @ W2[64,16] via WMMA f32 16x16x4
// =====================================================================
__global__ void gemm2_wmma(const float* __restrict__ emb,
                           const float* __restrict__ W2,
                           float* __restrict__ h2) {
  const int lane = threadIdx.x & 31;
  const int m    = lane & 15;
  const int kh   = lane >> 4;
  const int row0 = blockIdx.x * 16;

  v8f acc = {};
  const float* erow = emb + (size_t)(row0 + m) * 64;
  for (int k0 = 0; k0 < 64; k0 += 4) {
    const int ka = k0 + 2 * kh;
    v2f a;
    a.x = erow[ka];
    a.y = erow[ka + 1];
    v2f b;
    b.x = W2[(size_t)ka       * 16 + m];
    b.y = W2[(size_t)(ka + 1) * 16 + m];
    acc = __builtin_amdgcn_wmma_f32_16x16x4_f32(false, a, false, b,
                                                (short)0, acc, false, false);
  }
#pragma unroll
  for (int r = 0; r < 8; ++r) {
    const int row = row0 + r + 8 * kh;
    h2[(size_t)row * 16 + m] = acc[r];
  }
}

// ---- per-node attention scores, layer 1: [N,4] each --------------------
__global__ void att1_kernel(const float* __restrict__ h1,
                            const float* __restrict__ att_src,
                            const float* __restrict__ att_dst,
                            float* __restrict__ asrc, float* __restrict__ adst,
                            int N) {
  int tid = blockIdx.x * blockDim.x + threadIdx.x;   // (node, head)
  if (tid >= N * 4) return;
  int h = tid & 3, n = tid >> 2;
  const float* hp = h1 + (size_t)n * 64 + h * 16;
  float s = 0.f, t = 0.f;
#pragma unroll
  for (int c = 0; c < 16; ++c) {
    float v = hp[c];
    s += v * att_src[h * 16 + c];
    t += v * att_dst[h * 16 + c];
  }
  asrc[n * 4 + h] = s;
  adst[n * 4 + h] = t;
}

// ---- per-node attention scores, layer 2: [N] each ----------------------
__global__ void att2_kernel(const float* __restrict__ h2,
                            const float* __restrict__ att_src,
                            const float* __restrict__ att_dst,
                            float* __restrict__ asrc, float* __restrict__ adst,
                            int N) {
  int n = blockIdx.x * blockDim.x + threadIdx.x;
  if (n >= N) return;
  const float* hp = h2 + (size_t)n * 16;
  float s = 0.f, t = 0.f;
#pragma unroll
  for (int c = 0; c < 16; ++c) {
    float v = hp[c];
    s += v * att_src[c];
    t += v * att_dst[c];
  }
  asrc[n] = s;
  adst[n] = t;
}

// ---- segment max over incoming edges, layer 1 (edges incl. self-loops) -
__global__ void edge_max1(const int* __restrict__ ei,
                          const float* __restrict__ asrc,
                          const float* __restrict__ adst,
                          unsigned* __restrict__ mbuf, int E, int N) {
  int tid = blockIdx.x * blockDim.x + threadIdx.x;   // (edge, head)
  int total = E + N;
  int e = tid >> 2;
  if (e >= total) return;
  int h = tid & 3;
  int s, d;
  if (e < E) { s = ei[e]; d = ei[E + e]; } else { s = d = e - E; }
  float v = lrelu(asrc[s * 4 + h] + adst[d * 4 + h]);
  atomicMax(&mbuf[d * 4 + h], fenc(v));
}

// ---- one wave per edge: accumulate exp-weights and weighted messages ---
__global__ void edge_accum1(const int* __restrict__ ei,
                            const float* __restrict__ asrc,
                            const float* __restrict__ adst,
                            const unsigned* __restrict__ mbuf,
                            const float* __restrict__ h1,
                            float* __restrict__ denom,
                            float* __restrict__ acc, int E, int N) {
  int wid  = (blockIdx.x * blockDim.x + threadIdx.x) >> 5;
  int lane = threadIdx.x & 31;
  int total = E + N;
  if (wid >= total) return;
  int s, d;
  if (wid < E) { s = ei[wid]; d = ei[E + wid]; } else { s = d = wid - E; }
#pragma unroll
  for (int c = lane; c < 64; c += 32) {
    int h = c >> 4;
    float e  = lrelu(asrc[s * 4 + h] + adst[d * 4 + h]);
    float ex = expf(e - fdec(mbuf[d * 4 + h]));
    if ((c & 15) == 0) atomicAdd(&denom[d * 4 + h], ex);
    atomicAdd(&acc[(size_t)d * 64 + c], ex * h1[(size_t)s * 64 + c]);
  }
}

// ---- normalize + bias + ELU -> embeddings slice of d_out ---------------
__global__ void finish1(const float* __restrict__ acc,
                        const float* __restrict__ denom,
                        const float* __restrict__ b1,
                        float* __restrict__ emb, int N) {
  int tid = blockIdx.x * blockDim.x + threadIdx.x;
  if (tid >= N * 64) return;
  int c = tid & 63, n = tid >> 6, h = c >> 4;
  float v = acc[(size_t)n * 64 + c] / fmaxf(denom[n * 4 + h], 1e-16f) + b1[c];
  emb[(size_t)n * 64 + c] = v > 0.f ? v : expm1f(v);   // ELU
}

// ---- layer-2 edge max (single head) ------------------------------------
__global__ void edge_max2(const int* __restrict__ ei,
                          const float* __restrict__ asrc,
                          const float* __restrict__ adst,
                          unsigned* __restrict__ mbuf, int E, int N) {
  int e = blockIdx.x * blockDim.x + threadIdx.x;
  int total = E + N;
  if (e >= total) return;
  int s, d;
  if (e < E) { s = ei[e]; d = ei[E + e]; } else { s = d = e - E; }
  atomicMax(&mbuf[d], fenc(lrelu(asrc[s] + adst[d])));
}

// ---- layer-2 accumulate: 16 threads per edge ---------------------------
__global__ void edge_accum2(const int* __restrict__ ei,
                            const float* __restrict__ asrc,
                            const float* __restrict__ adst,
                            const unsigned* __restrict__ mbuf,
                            const float* __restrict__ h2,
                            float* __restrict__ denom,
                            float* __restrict__ acc, int E, int N) {
  int tid = blockIdx.x * blockDim.x + threadIdx.x;
  int e = tid >> 4;
  int total = E + N;
  if (e >= total) return;
  int c = tid & 15;
  int s, d;
  if (e < E) { s = ei[e]; d = ei[E + e]; } else { s = d = e - E; }
  float ev = lrelu(asrc[s] + adst[d]);
  float ex = expf(ev - fdec(mbuf[d]));
  if (c == 0) atomicAdd(&denom[d], ex);
  atomicAdd(&acc[(size_t)d * 16 + c], ex * h2[(size_t)s * 16 + c]);
}

// ---- normalize + bias + log_softmax -> logits slice of d_out -----------
__global__ void finish2(const float* __restrict__ acc,
                        const float* __restrict__ denom,
                        const float* __restrict__ b2,
                        float* __restrict__ logits, int N) {
  int n = blockIdx.x * blockDim.x + threadIdx.x;
  if (n >= N) return;
  float inv = 1.0f / fmaxf(denom[n], 1e-16f);
  float v[16];
  float mx = -3.4e38f;
#pragma unroll
  for (int c = 0; c < 16; ++c) {
    v[c] = acc[(size_t)n * 16 + c] * inv + b2[c];
    mx = fmaxf(mx, v[c]);
  }
  float sum = 0.f;
#pragma unroll
  for (int c = 0; c < 16; ++c) sum += expf(v[c] - mx);
  float lse = mx + logf(sum);
#pragma unroll
  for (int c = 0; c < 16; ++c) logits[(size_t)n * 16 + c] = v[c] - lse;
}

extern "C" void kernel_launch(void* const* d_in, const int* in_sizes, int n_in,
                              void* d_out, int out_size, void* d_ws, size_t ws_size,
                              hipStream_t stream) {
  const float* x        = (const float*)d_in[0];
  const int*   ei       = (const int*)d_in[1];
  const float* W1       = (const float*)d_in[2];
  const float* att_src1 = (const float*)d_in[3];
  const float* att_dst1 = (const float*)d_in[4];
  const float* b1       = (const float*)d_in[5];
  const float* W2       = (const float*)d_in[6];
  const float* att_src2 = (const float*)d_in[7];
  const float* att_dst2 = (const float*)d_in[8];
  const float* b2       = (const float*)d_in[9];

  const int N = in_sizes[0] / 128;   // 100000
  const int E = in_sizes[1] / 2;     // 1600000
  const int Etot = E + N;            // with self-loops

  // workspace carve-up (floats)
  float*    h1     = (float*)d_ws;              // N*64
  float*    asrc1  = h1     + (size_t)N * 64;   // N*4
  float*    adst1  = asrc1  + (size_t)N * 4;    // N*4
  unsigned* mbuf1  = (unsigned*)(adst1 + (size_t)N * 4); // N*4
  float*    denom1 = (float*)(mbuf1 + (size_t)N * 4);    // N*4
  float*    acc1   = denom1 + (size_t)N * 4;    // N*64
  float*    h2     = acc1   + (size_t)N * 64;   // N*16
  float*    asrc2  = h2     + (size_t)N * 16;   // N
  float*    adst2  = asrc2  + (size_t)N;        // N
  unsigned* mbuf2  = (unsigned*)(adst2 + (size_t)N);     // N
  float*    denom2 = (float*)(mbuf2 + (size_t)N);        // N
  float*    acc2   = denom2 + (size_t)N;        // N*16

  float* logits = (float*)d_out;                 // [N,16]
  float* emb    = logits + (size_t)N * 16;       // [N,64]

  // zero the atomic accumulators (mbuf1|denom1|acc1 and mbuf2|denom2|acc2
  // are each contiguous)
  hipMemsetAsync(mbuf1, 0, (size_t)N * 72 * sizeof(float), stream);
  hipMemsetAsync(mbuf2, 0, (size_t)N * 18 * sizeof(float), stream);

  // ---- layer 1 ----
  gemm1_wmma<<<N / 16, 128, 0, stream>>>(x, W1, h1);
  att1_kernel<<<(N * 4 + 255) / 256, 256, 0, stream>>>(h1, att_src1, att_dst1,
                                                       asrc1, adst1, N);
  edge_max1<<<((size_t)Etot * 4 + 255) / 256, 256, 0, stream>>>(ei, asrc1, adst1,
                                                                mbuf1, E, N);
  edge_accum1<<<((size_t)Etot * 32 + 255) / 256, 256, 0, stream>>>(
      ei, asrc1, adst1, mbuf1, h1, denom1, acc1, E, N);
  finish1<<<((size_t)N * 64 + 255) / 256, 256, 0, stream>>>(acc1, denom1, b1,
                                                            emb, N);

  // ---- layer 2 ----
  gemm2_wmma<<<N / 16, 32, 0, stream>>>(emb, W2, h2);
  att2_kernel<<<(N + 255) / 256, 256, 0, stream>>>(h2, att_src2, att_dst2,
                                                   asrc2, adst2, N);
  edge_max2<<<(Etot + 255) / 256, 256, 0, stream>>>(ei, asrc2, adst2, mbuf2, E, N);
  edge_accum2<<<((size_t)Etot * 16 + 255) / 256, 256, 0, stream>>>(
      ei, asrc2, adst2, mbuf2, h2, denom2, acc2, E, N);
  finish2<<<(N + 255) / 256, 256, 0, stream>>>(acc2, denom2, b2, logits, N);
}